// MCMCDA_15058155340472
// MI455X (gfx1250) — compile-verified
//
#include <hip/hip_runtime.h>

// ---------------------------------------------------------------------------
// MCMC + delayed-acceptance chain for MI455X (gfx1250, wave32).
// Single persistent workgroup runs the whole serial chain; 1024 obs points
// distributed 4-per-thread over 256 threads (8 waves). Wave reduction via
// V_WMMA_F32_16X16X4_F32 (B=ones -> row sums) + ds_swizzle SWAPX16; cross-wave
// via LDS ping-pong with a single barrier per reduction and a tree combine.
// RNG streams prefetched ahead with global_prefetch_b8 to hide HBM latency on
// the serial chain.
// ---------------------------------------------------------------------------

typedef float v2f __attribute__((ext_vector_type(2)));
typedef float v8f __attribute__((ext_vector_type(8)));

#define NPTS      1024
#define TPB       256
#define PPT       4                 // points per thread
#define NWAVES    (TPB / 32)
#define ITER_MCMC 100000
#define ITER_DA2  60000
#define NDA       20000
#define PF_DIST   64                // prefetch distance (iterations)
#define SWAPX16   0x401F            // ds_swizzle: and=0x1f, or=0, xor=0x10

__device__ __forceinline__ float wave_sum32(float v) {
#if __has_builtin(__builtin_amdgcn_wmma_f32_16x16x4_f32)
  // A (16x4): VGPR0 = partial (K=0 lanes 0-15, K=2 lanes 16-31), VGPR1 = 0.
  // B (4x16) = all ones  =>  D[m,n] = v[m] + v[m+16] for every n.
  v2f a; a[0] = v;    a[1] = 0.0f;
  v2f b; b[0] = 1.0f; b[1] = 1.0f;
  v8f c = {0.f, 0.f, 0.f, 0.f, 0.f, 0.f, 0.f, 0.f};
  v8f d = __builtin_amdgcn_wmma_f32_16x16x4_f32(
      /*neg_a=*/false, a, /*neg_b=*/false, b,
      /*c_mod=*/(short)0, c, /*reuse_a=*/false, /*reuse_b=*/false);
  // lane n(<16): sum_{m=0..7} (v[m]+v[m+16]); lane n+16: sum_{m=8..15}(...)
  float s = ((d[0] + d[1]) + (d[2] + d[3])) + ((d[4] + d[5]) + (d[6] + d[7]));
  // lane <-> lane^16 exchange, no address setup (group-of-32 swizzle).
  s += __int_as_float(
      __builtin_amdgcn_ds_swizzle(__float_as_int(s), SWAPX16));
  return s;
#else
  for (int m = 16; m >= 1; m >>= 1) v += __shfl_xor(v, m, 32);
  return v;
#endif
}

// Ping-pong LDS combine: one s_barrier per reduction is sufficient — a wave
// cannot re-write buffer `slot` (two calls later) before every wave has passed
// the *next* barrier, which happens only after all reads of `slot` completed.
__device__ __forceinline__ float block_sum(float v, float* red, int slot) {
  float w = wave_sum32(v);
  const int tid = threadIdx.x;
  if ((tid & 31) == 0) red[slot * NWAVES + (tid >> 5)] = w;
  __syncthreads();
  const float* r = red + slot * NWAVES;
  // pairwise tree: 3 dependent levels instead of a serial 8-add chain
  float s01 = r[0] + r[1];
  float s23 = r[2] + r[3];
  float s45 = r[4] + r[5];
  float s67 = r[6] + r[7];
  return (s01 + s23) + (s45 + s67);
}

// log posterior, coarse (NN surrogate) model.  inv_var = 1/0.25 -> -0.5*iv = -2
__device__ __forceinline__ float post_outer(float t0, float t1,
                                            const float* lx, const float* ly,
                                            const float* ob, float* pred,
                                            float* red, int slot) {
  float part = 0.0f;
#pragma unroll
  for (int j = 0; j < PPT; ++j) {
    float z = fmaf(lx[j], t0, ly[j] * t1);
    float s = __sinf(z);
    pred[j] = s;
    float e = ob[j] - s;
    part = fmaf(e, e, part);
  }
  float S = block_sum(part, red, slot);
  float prior = (fabsf(t0) <= 1.0f && fabsf(t1) <= 1.0f) ? 0.0f : -1e30f;
  return prior - 2.0f * S;
}

// log posterior, fine (solver) model: sin(z) + 0.1*cos(2z)
__device__ __forceinline__ float post_inner(float t0, float t1,
                                            const float* lx, const float* ly,
                                            const float* ob, float* pred,
                                            float* red, int slot) {
  float part = 0.0f;
#pragma unroll
  for (int j = 0; j < PPT; ++j) {
    float z = fmaf(lx[j], t0, ly[j] * t1);
    float s = __sinf(z) + 0.1f * __cosf(2.0f * z);
    pred[j] = s;
    float e = ob[j] - s;
    part = fmaf(e, e, part);
  }
  float S = block_sum(part, red, slot);
  float prior = (fabsf(t0) <= 1.0f && fabsf(t1) <= 1.0f) ? 0.0f : -1e30f;
  return prior - 2.0f * S;
}

__global__ void __launch_bounds__(TPB, 1)
chain_kernel(const float* __restrict__ locs, const float* __restrict__ obs,
             const float* __restrict__ theta0,
             const float* __restrict__ noise_mcmc, const float* __restrict__ u_mcmc,
             const float* __restrict__ noise_da, const float* __restrict__ u1_da,
             const float* __restrict__ u2_da,
             float* __restrict__ out_acc, float* __restrict__ out_th,
             float* __restrict__ out_lnn, float* __restrict__ out_lsol) {
  __shared__ float red[2 * NWAVES];
  const int tid = threadIdx.x;

  // Pin this thread's 4 observation points in registers.
  float lx[PPT], ly[PPT], ob[PPT];
#pragma unroll
  for (int j = 0; j < PPT; ++j) {
    int p = tid * PPT + j;
    lx[j] = locs[2 * p];
    ly[j] = locs[2 * p + 1];
    ob[j] = obs[p];
  }

  float t0 = theta0[0] * 2.0f - 1.0f;
  float t1 = theta0[1] * 2.0f - 1.0f;
  float dt = 0.1f;
  int   slot = 0;
  float scratch[PPT];

  // Cached log-posterior of current state (identical arithmetic to a fresh
  // recomputation, so trajectories match the reference).
  float lc = post_outer(t0, t1, lx, ly, ob, scratch, red, slot); slot ^= 1;

  // -------- Phase 1: adaptive random-walk Metropolis-Hastings ------------
  for (int i = 0; i < ITER_MCMC; ++i) {
    int pf = (i + PF_DIST < ITER_MCMC) ? (i + PF_DIST) : (ITER_MCMC - 1);
    __builtin_prefetch(&noise_mcmc[2 * pf], 0, 1);
    __builtin_prefetch(&u_mcmc[pf], 0, 1);

    float e0 = noise_mcmc[2 * i], e1 = noise_mcmc[2 * i + 1];
    float u  = u_mcmc[i];
    float p0 = fmaf(dt, e0, t0);
    float p1 = fmaf(dt, e1, t1);

    float lp = post_outer(p0, p1, lx, ly, ob, scratch, red, slot); slot ^= 1;
    float d  = lp - lc;
    float a  = __expf(fminf(d, 0.0f));
    if (u < a) { t0 = p0; t1 = p1; lc = lp; }
    dt = dt + dt * (a - 0.234f) / ((float)i + 1.0f);
  }

  // -------- Phase 2: delayed acceptance ----------------------------------
  float po[PPT], pi_[PPT];
  float lci = post_inner(t0, t1, lx, ly, ob, pi_, red, slot); slot ^= 1;
  int inner = 0;

  for (int i = 0; i < ITER_DA2; ++i) {
    int pf = (i + PF_DIST < ITER_DA2) ? (i + PF_DIST) : (ITER_DA2 - 1);
    __builtin_prefetch(&noise_da[2 * pf], 0, 1);
    __builtin_prefetch(&u1_da[pf], 0, 1);
    __builtin_prefetch(&u2_da[pf], 0, 1);

    float e0 = noise_da[2 * i], e1 = noise_da[2 * i + 1];
    float u1 = u1_da[i], u2 = u2_da[i];
    float p0 = fmaf(dt, e0, t0);
    float p1 = fmaf(dt, e1, t1);

    float lp = post_outer(p0, p1, lx, ly, ob, po, red, slot); slot ^= 1;
    float d     = lp - lc;
    float a     = __expf(fminf(d, 0.0f));
    float a_rec = __expf(fminf(-d, 0.0f));
    bool pass1  = (inner < NDA) && (u1 < a);

    // pass1 is block-uniform (all lanes computed identical scalars); ballot
    // forces a scalar branch so EXEC stays all-ones around the inner WMMAs.
    if (__ballot(pass1)) {
      float lpi = post_inner(p0, p1, lx, ly, ob, pi_, red, slot); slot ^= 1;
      int idx = inner;

      reinterpret_cast<float4*>(out_lnn  + (size_t)idx * NPTS)[tid] =
          make_float4(po[0], po[1], po[2], po[3]);
      reinterpret_cast<float4*>(out_lsol + (size_t)idx * NPTS)[tid] =
          make_float4(pi_[0], pi_[1], pi_[2], pi_[3]);

      float log_a2 = (lpi - lci) + __logf(a_rec + 1e-30f)
                                 - __logf(fmaxf(a, 1e-30f));
      float a2 = __expf(fminf(log_a2, 0.0f));
      bool acc2 = (u2 < a2);

      if (tid == 0) {
        out_th[2 * idx]     = p0;
        out_th[2 * idx + 1] = p1;
        out_acc[idx]        = acc2 ? 1.0f : 0.0f;
      }
      if (acc2) { t0 = p0; t1 = p1; lc = lp; lci = lpi; }
      ++inner;
    }
  }
}

// Zero-fill d_out (harness poisons it; unwritten acc/th_in/lnn/lsol tail
// entries must be 0 to match the reference's zero-initialized carries).
__global__ void zero_kernel(float* __restrict__ out, long long n) {
  long long n4 = n >> 2;
  float4* out4 = reinterpret_cast<float4*>(out);
  long long stride = (long long)gridDim.x * blockDim.x;
  for (long long i = (long long)blockIdx.x * blockDim.x + threadIdx.x;
       i < n4; i += stride)
    out4[i] = make_float4(0.f, 0.f, 0.f, 0.f);
  long long tail = n4 << 2;
  for (long long i = tail + (long long)blockIdx.x * blockDim.x + threadIdx.x;
       i < n; i += stride)
    out[i] = 0.0f;
}

extern "C" void kernel_launch(void* const* d_in, const int* in_sizes, int n_in,
                              void* d_out, int out_size, void* d_ws, size_t ws_size,
                              hipStream_t stream) {
  (void)in_sizes; (void)n_in; (void)d_ws; (void)ws_size;
  const float* locs       = (const float*)d_in[0];   // (1024,2)
  const float* obs        = (const float*)d_in[1];   // (1024,)
  const float* theta0     = (const float*)d_in[2];   // (2,)
  const float* noise_mcmc = (const float*)d_in[3];   // (100000,2)
  const float* u_mcmc     = (const float*)d_in[4];   // (100000,)
  const float* noise_da   = (const float*)d_in[5];   // (60000,2)
  const float* u1_da      = (const float*)d_in[6];   // (60000,)
  const float* u2_da      = (const float*)d_in[7];   // (60000,)

  float* out      = (float*)d_out;
  float* out_acc  = out;                              // (20000,)
  float* out_th   = out + NDA;                        // (20000,2)
  float* out_lnn  = out_th + 2 * (size_t)NDA;         // (20000,1024)
  float* out_lsol = out_lnn + (size_t)NDA * NPTS;     // (20000,1024)

  zero_kernel<<<4096, 256, 0, stream>>>(out, (long long)out_size);
  chain_kernel<<<1, TPB, 0, stream>>>(locs, obs, theta0, noise_mcmc, u_mcmc,
                                      noise_da, u1_da, u2_da,
                                      out_acc, out_th, out_lnn, out_lsol);
}